// HistogramLoss_41068477285138
// MI455X (gfx1250) — compile-verified
//
#include <hip/hip_runtime.h>

#define NUM_BINS 256

typedef __attribute__((ext_vector_type(2))) float v2f;
typedef __attribute__((ext_vector_type(8))) float v8f;

// ---------------------------------------------------------------------------
// Kernel 0: zero the global histogram table (2 x 256 u32) in workspace.
// ---------------------------------------------------------------------------
__global__ __launch_bounds__(512) void hist_zero_kernel(unsigned int* __restrict__ gbins) {
  int i = threadIdx.x;
  if (i < 2 * NUM_BINS) gbins[i] = 0u;
}

// ---------------------------------------------------------------------------
// Kernel 1: streaming histogram pass. float4 loads, LDS-privatized bins,
// global atomic flush. Memory-bound: ~100 MB @ 23.3 TB/s -> ~4.3 us floor.
// ---------------------------------------------------------------------------
__device__ __forceinline__ void bin_one(float x, unsigned int* __restrict__ h) {
  // np.histogram(range=(0,1)) semantics: floor(x*256), right edge inclusive,
  // out-of-range dropped. Truncation == floor for x >= 0.
  if (x >= 0.0f && x <= 1.0f) {
    int idx = (int)(x * 256.0f);
    idx = idx > (NUM_BINS - 1) ? (NUM_BINS - 1) : idx;
    atomicAdd(&h[idx], 1u);
  }
}

__global__ __launch_bounds__(256) void hist_accum_kernel(
    const float* __restrict__ in, const float* __restrict__ tg,
    unsigned int* __restrict__ gbins, int n4, int n) {
  __shared__ unsigned int lbin[2 * NUM_BINS];
  for (int i = threadIdx.x; i < 2 * NUM_BINS; i += blockDim.x) lbin[i] = 0u;
  __syncthreads();

  const float4* __restrict__ in4 = (const float4*)in;
  const float4* __restrict__ tg4 = (const float4*)tg;
  const int stride = gridDim.x * blockDim.x;

  for (int i = blockIdx.x * blockDim.x + threadIdx.x; i < n4; i += stride) {
    float4 a = in4[i];               // global_load_b128
    bin_one(a.x, lbin);
    bin_one(a.y, lbin);
    bin_one(a.z, lbin);
    bin_one(a.w, lbin);
    float4 b = tg4[i];               // global_load_b128
    bin_one(b.x, lbin + NUM_BINS);
    bin_one(b.y, lbin + NUM_BINS);
    bin_one(b.z, lbin + NUM_BINS);
    bin_one(b.w, lbin + NUM_BINS);
  }

  // Tail elements (n % 4), handled by one thread (n is 4-divisible here, but
  // stay correct for any n).
  if (blockIdx.x == 0 && threadIdx.x == 0) {
    for (int i = n4 * 4; i < n; ++i) {
      bin_one(in[i], lbin);
      bin_one(tg[i], lbin + NUM_BINS);
    }
  }

  __syncthreads();
  for (int i = threadIdx.x; i < 2 * NUM_BINS; i += blockDim.x) {
    unsigned int c = lbin[i];
    if (c) atomicAdd(&gbins[i], c);  // global_atomic_add_u32
  }
}

// ---------------------------------------------------------------------------
// Kernel 2: one wave (32 lanes). WMMA-based reductions with an all-ones
// A matrix: D[m][n] = sum_k B[k][n], so sum of all lanes of c[0] == 2 * total
// (rows 0 and 8 are identical), independent of B element placement.
// Counts < 2^24 -> all f32 accumulation exact.
// ---------------------------------------------------------------------------
__device__ __forceinline__ float wmma_total_x2(const float* v8) {
  // Reduce 8 per-lane f32 values (256 total across the wave) via 4 chained
  // V_WMMA_F32_16X16X4_F32 ops with A = ones(16x4).
  v2f ones; ones.x = 1.0f; ones.y = 1.0f;
  v8f acc = {};
  for (int t = 0; t < 4; ++t) {
    v2f b; b.x = v8[2 * t]; b.y = v8[2 * t + 1];
    acc = __builtin_amdgcn_wmma_f32_16x16x4_f32(
        /*neg_a=*/false, ones, /*neg_b=*/false, b,
        /*c_mod=*/(short)0, acc, /*reuse_a=*/false, /*reuse_b=*/false);
  }
  // Cross-lane sum of acc[0] over all 32 lanes = 2 * grand total.
  float s = acc[0];
  for (int off = 16; off >= 1; off >>= 1) s += __shfl_xor(s, off, 32);
  return s;
}

__global__ __launch_bounds__(32) void hist_final_kernel(
    const unsigned int* __restrict__ gbins, float* __restrict__ out) {
  const int lane = threadIdx.x;  // 0..31, single wave, EXEC all ones
  float ci[8], ct[8];
  for (int j = 0; j < 8; ++j) {
    ci[j] = (float)gbins[lane * 8 + j];
    ct[j] = (float)gbins[NUM_BINS + lane * 8 + j];
  }

  float si = 0.5f * wmma_total_x2(ci);   // total in-range input samples
  float st = 0.5f * wmma_total_x2(ct);   // total in-range target samples

  const float rsi = 1.0f / si;
  const float rst = 1.0f / st;
  float d[8];
  for (int j = 0; j < 8; ++j) d[j] = fabsf(ci[j] * rsi - ct[j] * rst);

  float l1 = 0.5f * wmma_total_x2(d);
  if (lane == 0) out[0] = l1;
}

// ---------------------------------------------------------------------------
extern "C" void kernel_launch(void* const* d_in, const int* in_sizes, int n_in,
                              void* d_out, int out_size, void* d_ws, size_t ws_size,
                              hipStream_t stream) {
  const float* in = (const float*)d_in[0];
  const float* tg = (const float*)d_in[1];
  float* out = (float*)d_out;
  unsigned int* gbins = (unsigned int*)d_ws;  // 2*256 u32 = 2 KB scratch

  const int n  = in_sizes[0];
  const int n4 = n >> 2;

  int blocks = 1024;  // 8192 wave32 waves: enough to saturate 23.3 TB/s HBM
  int needed = (n4 + 255) / 256;
  if (needed < 1) needed = 1;
  if (blocks > needed) blocks = needed;

  hist_zero_kernel<<<1, 512, 0, stream>>>(gbins);
  hist_accum_kernel<<<blocks, 256, 0, stream>>>(in, tg, gbins, n4, n);
  hist_final_kernel<<<1, 32, 0, stream>>>(gbins, out);
}